// TimeCaps_420906795152
// MI455X (gfx1250) — compile-verified
//
#include <hip/hip_runtime.h>
#include <hip/hip_bf16.h>

// ---------------------------------------------------------------------------
// Sizes from the reference
// ---------------------------------------------------------------------------
#define NB   64          // batch
#define LL   4500        // sequence length
#define KCH  64          // conv1 out channels
#define G1   7
#define G2   5
#define KDIM (KCH * G2)  // 320 = GEMM K for psi_m
#define NPOS 64          // positions per WMMA block chunk
#define TILE_W 68        // NPOS + halo(2+2)
#define AP   8
#define G3   3
#define NCH_ELEMS ((size_t)NB * KCH * LL)            // 18,432,000 floats
#define H4_ELEMS  ((size_t)NB * KCH * AP * LL)       // 147,456,000 floats

typedef __attribute__((ext_vector_type(2))) float v2f;
typedef __attribute__((ext_vector_type(8))) float v8f;

// gfx1250 async global->LDS copy (ASYNCcnt-tracked), with safe fallback.
#if defined(__has_builtin)
#  if __has_builtin(__builtin_amdgcn_global_load_async_to_lds_b32)
#    define HAVE_ASYNC_LDS 1
#  endif
#  if __has_builtin(__builtin_amdgcn_s_wait_asynccnt)
#    define HAVE_WAIT_ASYNC_BUILTIN 1
#  endif
#endif

#ifdef HAVE_ASYNC_LDS
typedef __attribute__((address_space(1))) int async_g_i32;  // global (AS1)
typedef __attribute__((address_space(3))) int async_l_i32;  // LDS (AS3)
static __device__ __forceinline__ void async_copy_b32(const float* g, float* lds) {
  __builtin_amdgcn_global_load_async_to_lds_b32((async_g_i32*)g, (async_l_i32*)lds,
                                                /*offset=*/0, /*cpol=*/0);
}
static __device__ __forceinline__ void async_wait0() {
#ifdef HAVE_WAIT_ASYNC_BUILTIN
  __builtin_amdgcn_s_wait_asynccnt(0);
#else
  asm volatile("s_wait_asynccnt 0" ::: "memory");
#endif
}
#endif

// ---------------------------------------------------------------------------
// Kernel 1: conv1d, 1 -> 64 channels, k=7, SAME (pad 3)
// grid (ceil(L/256), K, N), block 256
// ---------------------------------------------------------------------------
__global__ __launch_bounds__(256) void conv1_kernel(
    const float* __restrict__ x, const float* __restrict__ w1,
    float* __restrict__ h1) {
  const int l = blockIdx.x * 256 + threadIdx.x;
  const int k = blockIdx.y;
  const int n = blockIdx.z;
  if (l >= LL) return;
  const float* xr = x + (size_t)n * LL;
  const float* wr = w1 + k * G1;    // uniform per block -> scalar loads
  float s = 0.f;
#pragma unroll
  for (int g = 0; g < G1; ++g) {
    const int li = l + g - 3;
    const float xv = (li >= 0 && li < LL) ? xr[li] : 0.f;
    s = fmaf(xv, wr[g], s);
  }
  h1[((size_t)n * KCH + k) * LL + l] = s;
}

// ---------------------------------------------------------------------------
// Kernel 2: psi_m conv as WMMA GEMM.
//   out[n, co, l] = sum_{ci,g} h1[n, ci, l+g-2] * w[co, ci, g]
//   A = weights (M=64 x K=320), K ordered as (g, ci) inside the loop.
//   B = LDS-staged input patches (K x 64 positions), staged with
//       GLOBAL_LOAD_ASYNC_TO_LDS_B32 + s_wait_asynccnt when available.
//   V_WMMA_F32_16X16X4_F32, exact f32 accumulation.
// grid (ceil(L/64), N), block 256 = 8 waves:
//   wave = (m_tile in 0..3) x (n_group in 0..1); each wave owns two 16x16
//   output tiles and runs 5*16*2 = 160 WMMAs.
// ---------------------------------------------------------------------------
__global__ __launch_bounds__(256) void psi_m_wmma_kernel(
    const float* __restrict__ h1, const float* __restrict__ w,
    float* __restrict__ h2) {
  __shared__ float tile[KCH * TILE_W];

  const int tid = threadIdx.x;
  const int n  = blockIdx.y;
  const int l0 = blockIdx.x * NPOS;

  // Stage input tile (64 channels x 68 positions, halo of 2 each side).
#ifdef HAVE_ASYNC_LDS
  for (int idx = tid; idx < KCH * TILE_W; idx += 256) {
    const int ci = idx / TILE_W;
    const int ll = idx - ci * TILE_W;
    const int l  = l0 + ll - 2;
    if (l >= 0 && l < LL) {
      async_copy_b32(&h1[((size_t)n * KCH + ci) * LL + l], &tile[idx]);
    } else {
      tile[idx] = 0.f;               // zero halo via direct DS store
    }
  }
  async_wait0();
#else
  for (int idx = tid; idx < KCH * TILE_W; idx += 256) {
    const int ci = idx / TILE_W;
    const int ll = idx - ci * TILE_W;
    const int l  = l0 + ll - 2;
    tile[idx] = (l >= 0 && l < LL) ? h1[((size_t)n * KCH + ci) * LL + l] : 0.f;
  }
#endif
  __syncthreads();

  const int wv      = tid >> 5;
  const int lane    = tid & 31;
  const int m_base  = (wv & 3) * 16;
  const int nb0     = (wv >> 2) * 32;   // this wave's n-tiles: nb0, nb0+16
  const int lanecol = lane & 15;
  const int k_half  = lane >> 4;        // selects K pair within the 16x4 A frag

  const float* wrow = w + (size_t)(m_base + lanecol) * KDIM;  // row-major (co, ci, g)

  v8f acc0 = {};
  v8f acc1 = {};

#pragma unroll
  for (int g = 0; g < G2; ++g) {
#pragma unroll
    for (int cib = 0; cib < KCH; cib += 4) {
      // A fragment (16x4 f32): lanes 0-15 -> K={k0,k0+1}, lanes 16-31 -> K={k0+2,k0+3}
      // with K = g*64 + ci  ->  weight index = ci*5 + g.
      const int wi = (cib + 2 * k_half) * G2 + g;
      v2f a;
      a.x = wrow[wi];
      a.y = wrow[wi + G2];

      // B fragment (4x16 f32): row K striped across lane halves per VGPR.
      const int rb0 = (cib + k_half) * TILE_W + lanecol + g;      // rows k0 / k0+1
      const int rb1 = (cib + 2 + k_half) * TILE_W + lanecol + g;  // rows k0+2 / k0+3
      v2f b0, b1;
      b0.x = tile[rb0 + nb0];       b0.y = tile[rb1 + nb0];
      b1.x = tile[rb0 + nb0 + 16];  b1.y = tile[rb1 + nb0 + 16];

      acc0 = __builtin_amdgcn_wmma_f32_16x16x4_f32(false, a, false, b0,
                                                   (short)0, acc0, false, false);
      acc1 = __builtin_amdgcn_wmma_f32_16x16x4_f32(false, a, false, b1,
                                                   (short)0, acc1, false, false);
    }
  }

  // C/D layout: VGPR r, lanes 0-15 -> M = m_base + r, lanes 16-31 -> M = m_base + 8 + r.
  const int mrow = m_base + 8 * k_half;
#pragma unroll
  for (int r = 0; r < 8; ++r) {
    int l = l0 + nb0 + lanecol;
    if (l < LL) h2[((size_t)n * KCH + mrow + r) * LL + l] = acc0[r];
    l += 16;
    if (l < LL) h2[((size_t)n * KCH + mrow + r) * LL + l] = acc1[r];
  }
}

// ---------------------------------------------------------------------------
// Kernel 3: squash over L.  h <- h * 1/(1 + sum_l h^2)   (in place)
// grid NB*KCH, block 256
// ---------------------------------------------------------------------------
__global__ __launch_bounds__(256) void squash_kernel(float* __restrict__ h2) {
  __shared__ float red[256];
  const int tid = threadIdx.x;
  float* p = h2 + (size_t)blockIdx.x * LL;

  float s = 0.f;
  for (int l = tid; l < LL; l += 256) {
    const float v = p[l];
    s = fmaf(v, v, s);
  }
  red[tid] = s;
  __syncthreads();
  for (int st = 128; st > 0; st >>= 1) {
    if (tid < st) red[tid] += red[tid + st];
    __syncthreads();
  }
  const float scale = 1.f / (1.f + red[0]);
  for (int l = tid; l < LL; l += 256) p[l] *= scale;
}

// ---------------------------------------------------------------------------
// Kernel 4: psi_mA conv (stride 8 over H, k=(8,3), pad W=1) + bias
//   h4[n, co, p, l] = b[co] + sum_{a,g} h2[n, p*8+a, l+g-1] * w[co, a, g]
// grid (ceil(L/256), 8, N), block 256
// ---------------------------------------------------------------------------
__global__ __launch_bounds__(256) void psi_mA_kernel(
    const float* __restrict__ h2, const float* __restrict__ w,
    const float* __restrict__ bias, float* __restrict__ h4) {
  __shared__ float wS[KCH * AP * G3];   // 64*24
  __shared__ float bS[KCH];
  const int tid = threadIdx.x;
  for (int i = tid; i < KCH * AP * G3; i += 256) wS[i] = w[i];
  if (tid < KCH) bS[tid] = bias[tid];
  __syncthreads();

  const int l = blockIdx.x * 256 + tid;
  const int p = blockIdx.y;
  const int n = blockIdx.z;
  if (l >= LL) return;

  float inp[AP][G3];
#pragma unroll
  for (int a = 0; a < AP; ++a)
#pragma unroll
    for (int g = 0; g < G3; ++g) {
      const int li = l + g - 1;
      inp[a][g] = (li >= 0 && li < LL)
                      ? h2[((size_t)n * KCH + p * AP + a) * LL + li] : 0.f;
    }

  for (int co = 0; co < KCH; ++co) {
    float s = bS[co];
    const float* wr = &wS[co * AP * G3];
#pragma unroll
    for (int a = 0; a < AP; ++a)
#pragma unroll
      for (int g = 0; g < G3; ++g) s = fmaf(wr[a * G3 + g], inp[a][g], s);
    h4[(((size_t)n * KCH + co) * AP + p) * LL + l] = s;
  }
}

// ---------------------------------------------------------------------------
// Kernel 5: dynamic routing, N_ITERS=3 -> 2 iterations, last bj skipped.
//   iter0 (uniform c=1/8): one fused pass computes sqn and all 8
//   bj[i] = (1/(1+sqn)) * sum_l h[i][l]*sj[l]  via 9-way LDS tree reduction.
//   iter1: softmax(b) -> sqn pass -> vj = sj/(1+sqn) write pass.
// grid NB*KCH, block 256
// ---------------------------------------------------------------------------
__global__ __launch_bounds__(256) void routing_kernel(
    const float* __restrict__ h4, float* __restrict__ out) {
  __shared__ float part[9][257];
  const int tid = threadIdx.x;
  const int nc  = blockIdx.x;
  const float* base = h4 + (size_t)nc * AP * LL;

  // ---- iteration 0: c_i = 1/8 ----
  float sqn_p = 0.f;
  float t_p[AP];
#pragma unroll
  for (int i = 0; i < AP; ++i) t_p[i] = 0.f;

  for (int l = tid; l < LL; l += 256) {
    __builtin_prefetch(&base[l + 256], 0, 1);
    float hv[AP], sum = 0.f;
#pragma unroll
    for (int i = 0; i < AP; ++i) { hv[i] = base[i * LL + l]; sum += hv[i]; }
    const float sj = 0.125f * sum;
    sqn_p = fmaf(sj, sj, sqn_p);
#pragma unroll
    for (int i = 0; i < AP; ++i) t_p[i] = fmaf(hv[i], sj, t_p[i]);
  }
  part[0][tid] = sqn_p;
#pragma unroll
  for (int i = 0; i < AP; ++i) part[i + 1][tid] = t_p[i];
  __syncthreads();
  for (int st = 128; st > 0; st >>= 1) {
    if (tid < st) {
#pragma unroll
      for (int q = 0; q < 9; ++q) part[q][tid] += part[q][tid + st];
    }
    __syncthreads();
  }
  const float sqn0 = part[0][0];
  float b[AP];
#pragma unroll
  for (int i = 0; i < AP; ++i) b[i] = part[i + 1][0] / (1.f + sqn0);
  __syncthreads();   // before reusing `part`

  // ---- iteration 1: c = softmax(b) ----
  float bm = b[0];
#pragma unroll
  for (int i = 1; i < AP; ++i) bm = fmaxf(bm, b[i]);
  float c[AP], csum = 0.f;
#pragma unroll
  for (int i = 0; i < AP; ++i) { c[i] = __expf(b[i] - bm); csum += c[i]; }
  const float cinv = 1.f / csum;
#pragma unroll
  for (int i = 0; i < AP; ++i) c[i] *= cinv;

  sqn_p = 0.f;
  for (int l = tid; l < LL; l += 256) {
    __builtin_prefetch(&base[l + 256], 0, 1);
    float sj = 0.f;
#pragma unroll
    for (int i = 0; i < AP; ++i) sj = fmaf(c[i], base[i * LL + l], sj);
    sqn_p = fmaf(sj, sj, sqn_p);
  }
  part[0][tid] = sqn_p;
  __syncthreads();
  for (int st = 128; st > 0; st >>= 1) {
    if (tid < st) part[0][tid] += part[0][tid + st];
    __syncthreads();
  }
  const float inv = 1.f / (1.f + part[0][0]);

  // ---- write vj ----
  float* op = out + (size_t)nc * LL;
  for (int l = tid; l < LL; l += 256) {
    float sj = 0.f;
#pragma unroll
    for (int i = 0; i < AP; ++i) sj = fmaf(c[i], base[i * LL + l], sj);
    op[l] = sj * inv;
  }
}

// ---------------------------------------------------------------------------
// Host launcher. Workspace layout: [h1 | h2 | h4] (~737 MB total).
// ---------------------------------------------------------------------------
extern "C" void kernel_launch(void* const* d_in, const int* in_sizes, int n_in,
                              void* d_out, int out_size, void* d_ws, size_t ws_size,
                              hipStream_t stream) {
  (void)in_sizes; (void)n_in; (void)out_size; (void)ws_size;
  const float* x        = (const float*)d_in[0];  // (64,1,4500)
  const float* conv1_w  = (const float*)d_in[1];  // (64,1,7)
  const float* psi_m_w  = (const float*)d_in[2];  // (64,64,5)
  const float* psi_mA_w = (const float*)d_in[3];  // (64,1,8,3)
  const float* psi_mA_b = (const float*)d_in[4];  // (64,)
  float* out = (float*)d_out;                     // (64,8,36000) flat == (64,64,4500)

  float* h1 = (float*)d_ws;
  float* h2 = h1 + NCH_ELEMS;
  float* h4 = h2 + NCH_ELEMS;

  conv1_kernel     <<<dim3((LL + 255) / 256, KCH, NB), 256, 0, stream>>>(x, conv1_w, h1);
  psi_m_wmma_kernel<<<dim3((LL + NPOS - 1) / NPOS, NB), 256, 0, stream>>>(h1, psi_m_w, h2);
  squash_kernel    <<<NB * KCH, 256, 0, stream>>>(h2);
  psi_mA_kernel    <<<dim3((LL + 255) / 256, AP, NB), 256, 0, stream>>>(h2, psi_mA_w, psi_mA_b, h4);
  routing_kernel   <<<NB * KCH, 256, 0, stream>>>(h4, out);
}